// BilateralFilterDenoising_56573309223719
// MI455X (gfx1250) — compile-verified
//
#include <hip/hip_runtime.h>
#include <math.h>

// Bilateral filter denoising, 9x9 window, sigma_space=1.5, sigma_intensity=0.1,
// reflect padding, output clamped to [0,1].
//
// Strategy (MI455X / gfx1250):
//  - compute-bound on transcendentals -> exactly one v_exp_f32 per tap
//    (spatial gaussian folded into the exp2 argument as a literal constant)
//  - stage 40x40 halo tile in LDS via CDNA5 async global->LDS loads
//    (ASYNCcnt + s_wait_asynccnt), taps read from LDS (ds_load_b32)

#define WSZ   9
#define PADR  4
#define TILE  32
#define HALO  (TILE + 2 * PADR)      // 40
#define LDSW  (HALO + 1)             // 41 floats row stride (bank-conflict break)

// -50 * log2(e)  : intensity term scale (exp(-50 d^2) = exp2(-50*log2e * d^2))
#define K_INT  (-72.13475204444817f)
// -(1/4.5) * log2(e) : spatial term scale (exp(-r2/4.5) = exp2(K_SPC * r2))
#define K_SPC  (-0.32059889797532517f)

__global__ __launch_bounds__(256) void
BilateralFilterDenoising_56573309223719_kernel(const float* __restrict__ in,
                                               float* __restrict__ out,
                                               int H, int W)
{
    __shared__ float tile[HALO * LDSW];

    const int tid = threadIdx.x;          // 0..255
    const int bx  = blockIdx.x * TILE;
    const int by  = blockIdx.y * TILE;

    // ---- Phase 1: async load of 40x40 halo tile (reflect-padded) into LDS ----
    const unsigned lds_base = (unsigned)(uintptr_t)&tile[0]; // low 32 bits = LDS byte offset
    const int total = HALO * HALO;                           // 1600 elements

    for (int i = tid; i < total; i += 256) {
        const int ly = i / HALO;
        const int lx = i - ly * HALO;
        int gy = by + ly - PADR;
        int gx = bx + lx - PADR;
        // 'reflect' (no edge repeat): -1 -> 1, N -> N-2 (single reflection, PAD << N)
        gy = (gy < 0) ? -gy : ((gy >= H) ? (2 * H - 2 - gy) : gy);
        gx = (gx < 0) ? -gx : ((gx >= W) ? (2 * W - 2 - gx) : gx);

        const unsigned lds_off = lds_base + (unsigned)(ly * LDSW + lx) * 4u;
        const unsigned long long gaddr =
            (unsigned long long)(uintptr_t)(in + ((long long)gy * W + gx));

        // CDNA5 async copy: global memory -> LDS, tracked by ASYNCcnt.
        asm volatile("global_load_async_to_lds_b32 %0, %1, off"
                     :
                     : "v"(lds_off), "v"(gaddr)
                     : "memory");
    }

    // Wait for this wave's async copies, then sync the workgroup.
    asm volatile("s_wait_asynccnt 0" ::: "memory");
    __syncthreads();

    // ---- Phase 2: 81-tap bilateral filter from LDS ----
    const int tx  = tid & 31;         // 0..31
    const int ty0 = tid >> 5;         // 0..7

    for (int r = 0; r < 4; ++r) {
        const int ty = ty0 + r * 8;   // 0..31
        const int oy = by + ty;
        const int ox = bx + tx;

        const float c = tile[(ty + PADR) * LDSW + (tx + PADR)];
        float num = 0.0f;
        float den = 0.0f;

#pragma unroll
        for (int di = 0; di < WSZ; ++di) {
#pragma unroll
            for (int dj = 0; dj < WSZ; ++dj) {
                const float p = tile[(ty + di) * LDSW + (tx + dj)];
                const float d = p - c;
                // spatial constant folds to a literal per (di,dj) after unroll
                const float r2  = (float)((di - PADR) * (di - PADR) +
                                          (dj - PADR) * (dj - PADR));
                const float arg = fmaf(d * d, K_INT, K_SPC * r2);
                const float w   = __builtin_amdgcn_exp2f(arg);   // v_exp_f32
                num = fmaf(w, p, num);
                den += w;
            }
        }

        if (oy < H && ox < W) {
            const float v = num * __builtin_amdgcn_rcpf(den + 1e-10f);
            out[(long long)oy * W + ox] = __builtin_amdgcn_fmed3f(v, 0.0f, 1.0f);
        }
    }
}

extern "C" void kernel_launch(void* const* d_in, const int* in_sizes, int n_in,
                              void* d_out, int out_size, void* d_ws, size_t ws_size,
                              hipStream_t stream)
{
    (void)n_in; (void)d_ws; (void)ws_size; (void)out_size;

    const float* noisy = (const float*)d_in[0];
    float* out = (float*)d_out;

    // Input is (1,1,H,W) square fp32; derive side length.
    const long long n = (long long)in_sizes[0];
    int side = (int)llround(sqrt((double)n));
    if ((long long)side * side != n) side = 2048;  // reference default
    const int H = side, W = side;

    dim3 block(256, 1, 1);
    dim3 grid((W + TILE - 1) / TILE, (H + TILE - 1) / TILE, 1);
    BilateralFilterDenoising_56573309223719_kernel<<<grid, block, 0, stream>>>(noisy, out, H, W);
}